// res_GAT_block_5738076307725
// MI455X (gfx1250) — compile-verified
//
#include <hip/hip_runtime.h>
#include <hip/hip_bf16.h>

// ---------------------------------------------------------------------------
// res_GAT_block for MI455X (gfx1250, wave32).
// B=4, C=128, N=4096, K=8.
//
// Algebraic simplifications vs reference:
//   * softmax weights sum to 1  =>  out = (sum of 9 feats)/9 + act   (W unused)
//   * grouped_nn = gather(conv2_w @ act, idx)   (8x fewer FLOPs)
// KNN ranking: pd[n,m] ordering == (2*dot(n,m) - xx[m]) ordering (row const
// dropped); self is always rank-0. fp32 WMMA keeps ordering faithful.
// ---------------------------------------------------------------------------

#define BB 4
#define CC 128
#define NN 4096
#define KK 8

typedef __attribute__((ext_vector_type(2))) float v2f;
typedef __attribute__((ext_vector_type(8))) float v8f;

// ---------------------------------------------------------------- act/center
__global__ __launch_bounds__(256) void act_kernel(const float* __restrict__ points,
                                                  float* __restrict__ center) {
  size_t i = (size_t)blockIdx.x * 256 + threadIdx.x;   // B*C*N elements
  float p = points[i];
  center[i] = (p >= 0.0f) ? p : 0.01f * p;
}

// ---------------------------------------------------------------- row norms
__global__ __launch_bounds__(256) void xx_kernel(const float* __restrict__ points,
                                                 float* __restrict__ xx) {
  int i = blockIdx.x * 256 + threadIdx.x;              // b*N + n, total B*N
  int b = i >> 12;
  int n = i & (NN - 1);
  const float* P = points + (size_t)b * CC * NN + n;
  float s = 0.0f;
  #pragma unroll 8
  for (int c = 0; c < CC; ++c) {
    float v = P[(size_t)c * NN];
    s += v * v;
  }
  xx[i] = s;
}

// ---------------------------------------------------------------- KNN top-8
// One wave (32 threads) per 32 rows of one batch. Scores via f32 WMMA.
__global__ __launch_bounds__(32) void knn_kernel(const float* __restrict__ points,
                                                 const float* __restrict__ xx,
                                                 int* __restrict__ idxout) {
  __shared__ float At[32][CC + 2];   // [n_local][c], stride 130 (even, 8B-aligned, no bank conflicts)
  __shared__ float sc[32][17];       // score tile [n_local][m_local]

  const int lane = threadIdx.x;
  const int b    = blockIdx.y;
  const int n0   = blockIdx.x * 32;
  const float* P = points + (size_t)b * CC * NN;

  // Stage A tile: At[n][c] = P[c][n0+n]  (coalesced over n)
  for (int c = 0; c < CC; ++c)
    At[lane][c] = P[(size_t)c * NN + n0 + lane];
  __syncthreads();

  // Per-lane top-8 (lane owns row n0+lane), sorted descending, stable.
  float vv[KK];
  int   ii[KK];
  #pragma unroll
  for (int j = 0; j < KK; ++j) { vv[j] = -__builtin_inff(); ii[j] = 0; }

  const int  col  = lane & 15;
  const bool hi   = lane >= 16;
  const int  koff = hi ? 2 : 0;   // A 16x4 f32 layout: lanes 0-15 -> K0/K1, 16-31 -> K2/K3
  const int  r    = lane & 15;

  for (int m0 = 0; m0 < NN; m0 += 16) {
    v8f acc0 = {};  // rows n0+0..15
    v8f acc1 = {};  // rows n0+16..31
    for (int kb = 0; kb < CC; kb += 4) {
      v2f a0 = *(const v2f*)&At[r][kb + koff];
      v2f a1 = *(const v2f*)&At[r + 16][kb + koff];
      // B 4x16 f32: mirrored layout, lanes 0-15 -> rows K0/K1, 16-31 -> K2/K3
      v2f bf = { P[(size_t)(kb + koff) * NN + m0 + col],
                 P[(size_t)(kb + koff + 1) * NN + m0 + col] };
      acc0 = __builtin_amdgcn_wmma_f32_16x16x4_f32(false, a0, false, bf,
                                                   (short)0, acc0, false, false);
      acc1 = __builtin_amdgcn_wmma_f32_16x16x4_f32(false, a1, false, bf,
                                                   (short)0, acc1, false, false);
    }
    float xxm = xx[b * NN + m0 + col];
    const int rbase = hi ? 8 : 0;   // C layout: VGPR v -> rows v (lanes 0-15) / v+8 (16-31)
    #pragma unroll
    for (int v = 0; v < 8; ++v) {
      sc[rbase + v][col]      = 2.0f * acc0[v] - xxm;
      sc[16 + rbase + v][col] = 2.0f * acc1[v] - xxm;
    }
    __syncthreads();
    // Sequential ascending-m scan => jax top_k tie-break (lower index first).
    #pragma unroll 1
    for (int mm = 0; mm < 16; ++mm) {
      float val = sc[lane][mm];
      if (val > vv[KK - 1]) {
        int rank = 0;
        #pragma unroll
        for (int j = 0; j < KK; ++j) rank += (vv[j] >= val) ? 1 : 0;
        #pragma unroll
        for (int j = KK - 1; j >= 1; --j) {
          bool sh = (j > rank);
          vv[j] = sh ? vv[j - 1] : vv[j];
          ii[j] = sh ? ii[j - 1] : ii[j];
        }
        #pragma unroll
        for (int j = 0; j < KK; ++j)
          if (j == rank) { vv[j] = val; ii[j] = m0 + mm; }
      }
    }
    __syncthreads();
  }

  #pragma unroll
  for (int j = 0; j < KK; ++j)
    idxout[((size_t)b * NN + n0 + lane) * KK + j] = ii[j];
}

// ---------------------------------------------------------------- convact = conv2_w @ act
// One wave per 16(o) x 16(n) tile, f32 WMMA, K=128.
__global__ __launch_bounds__(32) void conv_kernel(const float* __restrict__ w,
                                                  const float* __restrict__ act,
                                                  float* __restrict__ convact) {
  const int lane = threadIdx.x;
  const int b  = blockIdx.z;
  const int n0 = blockIdx.x * 16;
  const int o0 = blockIdx.y * 16;
  const float* A = act + (size_t)b * CC * NN;

  const int  r    = lane & 15;
  const bool hi   = lane >= 16;
  const int  koff = hi ? 2 : 0;

  v8f acc = {};
  for (int kb = 0; kb < CC; kb += 4) {
    v2f af = *(const v2f*)&w[(size_t)(o0 + r) * CC + kb + koff];   // row-major, 8B aligned
    v2f bf = { A[(size_t)(kb + koff) * NN + n0 + r],
               A[(size_t)(kb + koff + 1) * NN + n0 + r] };
    acc = __builtin_amdgcn_wmma_f32_16x16x4_f32(false, af, false, bf,
                                                (short)0, acc, false, false);
  }
  const int rbase = hi ? 8 : 0;
  float* out = convact + (size_t)b * CC * NN;
  #pragma unroll
  for (int v = 0; v < 8; ++v)
    out[(size_t)(o0 + rbase + v) * NN + n0 + r] = acc[v];
}

// ---------------------------------------------------------------- gather + out
// Block 256 threads = 32 n-values x 8 neighbors. Loops over c.
__global__ __launch_bounds__(256) void gather_kernel(const float* __restrict__ act,
                                                     const float* __restrict__ convact,
                                                     const int* __restrict__ idxin,
                                                     float* __restrict__ grouped,
                                                     float* __restrict__ grouped_nn,
                                                     float* __restrict__ outp) {
  const int b  = blockIdx.y;
  const int n0 = blockIdx.x * 32;
  const int t  = threadIdx.x;
  const int nl = t >> 3;       // 0..31
  const int j  = t & 7;        // 0..7
  const int n  = n0 + nl;

  const int ind = idxin[((size_t)b * NN + n) * KK + j];
  const float* A  = act     + (size_t)b * CC * NN;
  const float* CA = convact + (size_t)b * CC * NN;

  for (int c = 0; c < CC; ++c) {
    float g  = A [(size_t)c * NN + ind];
    float gn = CA[(size_t)c * NN + ind];
    size_t base = ((size_t)b * CC + c) * ((size_t)NN * KK) + (size_t)n * KK + j;
    grouped[base]    = g;     // coalesced: block writes 256 consecutive floats
    grouped_nn[base] = gn;
    // sum over the 8 neighbors (groups of 8 lanes inside a wave32)
    float s = g;
    s += __shfl_xor(s, 1, 32);
    s += __shfl_xor(s, 2, 32);
    s += __shfl_xor(s, 4, 32);
    if (j == 0) {
      float a = A[(size_t)c * NN + n];
      outp[((size_t)b * CC + c) * NN + n] = (s + a) * (1.0f / 9.0f) + a;
    }
  }
}

// ---------------------------------------------------------------------------
extern "C" void kernel_launch(void* const* d_in, const int* in_sizes, int n_in,
                              void* d_out, int out_size, void* d_ws, size_t ws_size,
                              hipStream_t stream) {
  const float* points  = (const float*)d_in[0];
  // d_in[1] = W : unused (softmax weights cancel in `out`)
  const float* conv2_w = (const float*)d_in[2];
  // d_in[3] = k (=8) : hardcoded

  float* out_ptr    = (float*)d_out;                                  // (B,C,N)
  float* center_ptr = out_ptr + (size_t)BB * CC * NN;                 // (B,C,N,1) == act storage
  float* grouped    = center_ptr + (size_t)BB * CC * NN;              // (B,C,N,8)
  float* grouped_nn = grouped + (size_t)BB * CC * NN * KK;            // (B,C,N,8)

  // workspace layout
  float* xx      = (float*)d_ws;                                     // B*N
  int*   idx     = (int*)(xx + (size_t)BB * NN);                     // B*N*8
  float* convact = (float*)(idx + (size_t)BB * NN * KK);             // B*C*N

  act_kernel<<<dim3((BB * CC * NN) / 256), dim3(256), 0, stream>>>(points, center_ptr);
  xx_kernel<<<dim3((BB * NN) / 256), dim3(256), 0, stream>>>(points, xx);
  knn_kernel<<<dim3(NN / 32, BB), dim3(32), 0, stream>>>(points, xx, idx);
  conv_kernel<<<dim3(NN / 16, CC / 16, BB), dim3(32), 0, stream>>>(conv2_w, center_ptr, convact);
  gather_kernel<<<dim3(NN / 32, BB), dim3(256), 0, stream>>>(center_ptr, convact, idx,
                                                             grouped, grouped_nn, out_ptr);
}